// GraphMultiConvolution_3023656976524
// MI455X (gfx1250) — compile-verified
//
#include <hip/hip_runtime.h>
#include <hip/hip_bf16.h>

#define FEAT   256
#define KEXP   8
#define TILE_M 32                 // per-block node tile (2 stacked 16-row WMMA tiles/wave)
#define LDSS   (FEAT + 4)         // padded LDS row stride (floats) -> conflict-free A reads

typedef __attribute__((ext_vector_type(2))) float v2f;
typedef __attribute__((ext_vector_type(8))) float v8f;

// ---------------------------------------------------------------------------
// Kernel 1: zero the hi accumulator in workspace (float4 grid-stride).
// ---------------------------------------------------------------------------
__global__ void __launch_bounds__(256) zero_f4(float4* __restrict__ p, long n4) {
    long i = (long)blockIdx.x * blockDim.x + threadIdx.x;
    long stride = (long)gridDim.x * blockDim.x;
    float4 zv; zv.x = 0.f; zv.y = 0.f; zv.z = 0.f; zv.w = 0.f;
    for (; i < n4; i += stride) p[i] = zv;
}

// ---------------------------------------------------------------------------
// Kernel 2: SpMM scatter. 64 threads per edge, each thread handles 4 floats
// (float4 gather of x[col], 4 global f32 atomic adds into hi[row]).
// x (64MB) and hi (64MB) both sit in the 192MB L2, so gather/atomic traffic
// stays on-chip; HBM sees only ~150MB (edge lists + cold misses).
// ---------------------------------------------------------------------------
__global__ void __launch_bounds__(256) spmm_scatter(
    const float* __restrict__ x, const int* __restrict__ rows,
    const int* __restrict__ cols, const float* __restrict__ vals,
    float* __restrict__ hi, int E)
{
    long t = (long)blockIdx.x * blockDim.x + threadIdx.x;
    int e = (int)(t >> 6);            // 64 threads per edge
    if (e >= E) return;
    int j = (int)(t & 63);            // float4 slot within the 256-float row
    int r = rows[e];
    int c = cols[e];
    float v = vals[e];
    float4 xv = ((const float4*)(x + (long)c * FEAT))[j];
    float* dst = hi + (long)r * FEAT + j * 4;
    atomicAdd(dst + 0, v * xv.x);
    atomicAdd(dst + 1, v * xv.y);
    atomicAdd(dst + 2, v * xv.z);
    atomicAdd(dst + 3, v * xv.w);
}

// ---------------------------------------------------------------------------
// Kernel 3: fused K-expert combine with fp32 WMMA.
//   out[n,o] = sum_k z[n,k] * (hi[n,:] @ W[k][:,o]) + x[n,o]
// Block = 256 threads = 8 wave32. Block tile: 32 nodes x 128 cols.
// Each wave owns a 32x16 tile = TWO 16x16 accumulators that SHARE every B
// fragment (register-level B reuse -> 1 global load per WMMA instead of 2).
// Per expert: 2 x 64 V_WMMA_F32_16X16X4_F32 over the 256-deep reduction,
// then the per-expert 16x16 results are row-scaled by z[n,k] and accumulated.
// Full fp32 throughout -> numerically faithful to the reference.
// ---------------------------------------------------------------------------
__global__ void __launch_bounds__(256) moe_gemm(
    const float* __restrict__ hi,   // [N, FEAT] (workspace)
    const float* __restrict__ x,    // [N, FEAT] residual
    const float* __restrict__ z,    // [N, KEXP]
    const float* __restrict__ w,    // [KEXP, FEAT, FEAT]
    float* __restrict__ out)        // [N, FEAT]
{
    __shared__ float sh_hi[TILE_M * LDSS];
    __shared__ float sh_z[TILE_M * KEXP];

    const int m0   = blockIdx.x * TILE_M;     // node-tile base
    const int tid  = threadIdx.x;
    const int wave = tid >> 5;
    const int lane = tid & 31;

    // Stage hi tile: 32 rows x 256 floats = 2048 float4, 8 per thread.
    #pragma unroll
    for (int r = 0; r < 8; ++r) {
        int idx = tid + r * 256;              // 0..2047
        int row = idx >> 6;                   // 64 float4 per row
        int c4  = idx & 63;
        float4 v = ((const float4*)(hi + (long)(m0 + row) * FEAT))[c4];
        float* dst = &sh_hi[row * LDSS + c4 * 4];
        dst[0] = v.x; dst[1] = v.y; dst[2] = v.z; dst[3] = v.w;
    }
    // Stage z tile: 32 x 8 = 256 entries, one per thread.
    sh_z[tid] = z[(long)(m0 + (tid >> 3)) * KEXP + (tid & 7)];
    __syncthreads();

    const int o0       = blockIdx.y * 128 + wave * 16;  // this wave's 16 cols
    const int l15      = lane & 15;   // A: row M ; B/D: col N
    const int laneHalf = lane >> 4;   // selects K sub-pair / D row half

    v8f acc0 = {0.f, 0.f, 0.f, 0.f, 0.f, 0.f, 0.f, 0.f};
    v8f acc1 = {0.f, 0.f, 0.f, 0.f, 0.f, 0.f, 0.f, 0.f};

    for (int k = 0; k < KEXP; ++k) {
        v8f dk0 = {0.f, 0.f, 0.f, 0.f, 0.f, 0.f, 0.f, 0.f};
        v8f dk1 = {0.f, 0.f, 0.f, 0.f, 0.f, 0.f, 0.f, 0.f};
        // B base for this expert/column: W[k, kk, o0 + l15]
        const float* wk  = w + (long)k * FEAT * FEAT + o0 + l15;
        const float* ha0 = &sh_hi[l15 * LDSS];          // rows m0..m0+15
        const float* ha1 = &sh_hi[(l15 + 16) * LDSS];   // rows m0+16..m0+31

        #pragma unroll 8
        for (int i = 0; i < FEAT; i += 4) {
            const int kk = i + 2 * laneHalf;
            v2f b, a0, a1;
            b.x  = wk[(long)kk * FEAT];
            b.y  = wk[(long)(kk + 1) * FEAT];
            a0.x = ha0[kk];
            a0.y = ha0[kk + 1];
            a1.x = ha1[kk];
            a1.y = ha1[kk + 1];
            dk0 = __builtin_amdgcn_wmma_f32_16x16x4_f32(
                    false, a0, false, b, (short)0, dk0, false, false);
            dk1 = __builtin_amdgcn_wmma_f32_16x16x4_f32(
                    false, a1, false, b, (short)0, dk1, false, false);
        }
        // Row-scale the per-expert 16x16 results by z[n,k] and accumulate.
        // D layout: vgpr r holds row (r + 8*laneHalf), col l15.
        #pragma unroll
        for (int r = 0; r < 8; ++r) {
            const int rr = r + 8 * laneHalf;
            acc0[r] += dk0[r] * sh_z[rr * KEXP + k];
            acc1[r] += dk1[r] * sh_z[(rr + 16) * KEXP + k];
        }
    }

    // Epilogue: residual add + store (two row blocks).
    #pragma unroll
    for (int r = 0; r < 8; ++r) {
        const long row0 = (long)(m0 + r + 8 * laneHalf);
        const long idx0 = row0 * FEAT + o0 + l15;
        out[idx0] = acc0[r] + x[idx0];
        const long idx1 = idx0 + 16L * FEAT;
        out[idx1] = acc1[r] + x[idx1];
    }
}

// ---------------------------------------------------------------------------
// Host launcher.
// Inputs (setup_inputs order): 0=x [N,F], 1=h0 (unused), 2=z [N,K],
// 3=vals [E], 4=weights [K,F,F], 5=rows [E], 6=cols [E]. Output: [N,F] f32.
// Workspace: hi accumulator, N*F*4 = 64 MB.
// ---------------------------------------------------------------------------
extern "C" void kernel_launch(void* const* d_in, const int* in_sizes, int n_in,
                              void* d_out, int out_size, void* d_ws, size_t ws_size,
                              hipStream_t stream) {
    const float* x    = (const float*)d_in[0];
    const float* z    = (const float*)d_in[2];
    const float* vals = (const float*)d_in[3];
    const float* w    = (const float*)d_in[4];
    const int*   rows = (const int*)d_in[5];
    const int*   cols = (const int*)d_in[6];
    float* out = (float*)d_out;
    float* hi  = (float*)d_ws;

    const int N = in_sizes[0] / FEAT;   // 65536
    const int E = in_sizes[3];          // 2097152

    // 1) hi = 0
    long n4 = (long)N * FEAT / 4;
    zero_f4<<<2048, 256, 0, stream>>>((float4*)hi, n4);

    // 2) hi += scatter(vals * x[cols]) -> rows
    long threads = (long)E * 64;
    int blocks = (int)((threads + 255) / 256);
    spmm_scatter<<<blocks, 256, 0, stream>>>(x, rows, cols, vals, hi, E);

    // 3) out = sum_k z_k * (hi @ W_k) + x   (fp32 WMMA, 32x16 tiles/wave)
    dim3 grid(N / TILE_M, 2);
    moe_gemm<<<grid, 256, 0, stream>>>(hi, x, z, w, out);
}